// MoonEmbedding_29609504538698
// MI455X (gfx1250) — compile-verified
//
#include <hip/hip_runtime.h>
#include <hip/hip_bf16.h>

#define BATCH 1024
#define NE 32
#define NUP 16
#define NI 8
#define H0C 64
#define H1C 32
#define DEE 32
#define DC 64
#define DOUT 256

typedef float v2f __attribute__((ext_vector_type(2)));
typedef float v8f __attribute__((ext_vector_type(8)));

__device__ __forceinline__ float silu_f(float x) { return x / (1.0f + __expf(-x)); }
__device__ __forceinline__ float softplus_f(float x) { return (x > 20.0f) ? x : log1pf(__expf(x)); }

__device__ __forceinline__ v8f wmma4(v8f acc, float a0, float a1, float b0, float b1) {
  v2f a; a.x = a0; a.y = a1;
  v2f b; b.x = b0; b.y = b1;
  // D = A(16x4) * B(4x16) + C, fp32 WMMA
  return __builtin_amdgcn_wmma_f32_16x16x4_f32(false, a, false, b, (short)0, acc, false, false);
}

// 16x16 output tile accumulate, A rows m0..m0+15 in LDS (stride lda), B row-major global (ldb)
__device__ __forceinline__ v8f gemm_ldsA(const float* A, int lda, int m0,
                                         const float* __restrict__ Bw, int ldb, int n0,
                                         int K, int lane) {
  const int half = lane >> 4, idx = lane & 15;
  v8f acc = {0.f, 0.f, 0.f, 0.f, 0.f, 0.f, 0.f, 0.f};
  for (int k = 0; k < K; k += 4) {
    float a0 = A[(m0 + idx) * lda + k + 2 * half + 0];
    float a1 = A[(m0 + idx) * lda + k + 2 * half + 1];
    float b0 = Bw[(k + 2 * half + 0) * ldb + n0 + idx];
    float b1 = Bw[(k + 2 * half + 1) * ldb + n0 + idx];
    acc = wmma4(acc, a0, a1, b0, b1);
  }
  return acc;
}

// Same but A rows are contiguous in global memory (row stride lda)
__device__ __forceinline__ v8f gemm_globA(const float* __restrict__ A, int lda,
                                          const float* __restrict__ Bw, int ldb, int n0,
                                          int K, int lane) {
  const int half = lane >> 4, idx = lane & 15;
  v8f acc = {0.f, 0.f, 0.f, 0.f, 0.f, 0.f, 0.f, 0.f};
  for (int k = 0; k < K; k += 4) {
    float a0 = A[idx * lda + k + 2 * half + 0];
    float a1 = A[idx * lda + k + 2 * half + 1];
    float b0 = Bw[(k + 2 * half + 0) * ldb + n0 + idx];
    float b1 = Bw[(k + 2 * half + 1) * ldb + n0 + idx];
    acc = wmma4(acc, a0, a1, b0, b1);
  }
  return acc;
}

// ---------------- Kernel 1: ion embedding + ion-ion message passing ----------------
__global__ __launch_bounds__(128) void k_ion_embed(
    const float* __restrict__ diff_ii, const float* __restrict__ dist_ii,
    const float* __restrict__ feat_ion,
    const float* __restrict__ emb_w, const float* __restrict__ emb_b,
    const float* __restrict__ sb0, const float* __restrict__ sb1,
    const float* __restrict__ sscale, const float* __restrict__ sw0,
    const float* __restrict__ sw1, const float* __restrict__ swenv,
    const float* __restrict__ Gii,
    float* __restrict__ h_ion_0) {
  __shared__ float Hs[64 * 65];
  __shared__ float Es[64 * 65];
  __shared__ float Ys[64 * 33];
  __shared__ float Ws[64 * 33];
  __shared__ float hemb[8 * 64];
  const int b = blockIdx.x;
  const int tid = threadIdx.x;
  const int lane = tid & 31, wave = tid >> 5;
  const int half = lane >> 4, idx = lane & 15;

  for (int o = tid; o < NI * DC; o += 128) {
    int e = o >> 6, c = o & 63;
    float acc = emb_b[c];
    const float* f = feat_ion + (b * NI + e) * 8;
#pragma unroll
    for (int k = 0; k < 8; k++) acc += f[k] * emb_w[k * DC + c];
    hemb[o] = acc;
  }
  for (int o = tid; o < 64 * 64; o += 128) {
    int r = o >> 6, c = o & 63;
    int i = r >> 3, j = r & 7;
    const float* d = diff_ii + ((long)(b * NI + i) * NI + j) * 3;
    float dist = dist_ii[(long)(b * NI + i) * NI + j];
    float h = d[0] * sw0[c] + d[1] * sw0[64 + c] + d[2] * sw0[128 + c] + sb0[c];
    Hs[r * 65 + c] = silu_f(h);
    float sc = softplus_f(sscale[c]);
    float t = dist * sc;
    Es[r * 65 + c] = __expf(-t * t);
  }
  __syncthreads();
  for (int q = 0; q < 4; q++) {
    int job = wave * 4 + q;
    int mt = job >> 2, nt = (job >> 1) & 1, mat = job & 1;
    const float* A = mat ? Es : Hs;
    const float* Bw = mat ? swenv : sw1;
    v8f acc = gemm_ldsA(A, 65, mt * 16, Bw, H1C, nt * 16, H0C, lane);
    float* Dst = mat ? Ws : Ys;
#pragma unroll
    for (int r = 0; r < 8; r++)
      Dst[(mt * 16 + half * 8 + r) * 33 + nt * 16 + idx] = acc[r];
  }
  __syncthreads();
  for (int o = tid; o < 64 * 32; o += 128) {
    int r = o >> 5, c = o & 31;
    Ys[r * 33 + c] = (Ys[r * 33 + c] + sb1[c]) * Ws[r * 33 + c];
  }
  __syncthreads();
  for (int q = 0; q < 4; q++) {
    int job = wave * 4 + q;
    int mt = job >> 2, nt = job & 3;
    v8f acc = gemm_ldsA(Ys, 33, mt * 16, Gii, DC, nt * 16, H1C, lane);
#pragma unroll
    for (int r = 0; r < 8; r++)
      Hs[(mt * 16 + half * 8 + r) * 65 + nt * 16 + idx] = acc[r];  // gamma reuses Hs
  }
  __syncthreads();
  for (int o = tid; o < NI * DC; o += 128) {
    int i = o >> 6, c = o & 63;
    float acc = 0.f;
#pragma unroll
    for (int j = 0; j < 8; j++)
      acc += hemb[j * 64 + c] * Hs[(i * 8 + j) * 65 + c];
    h_ion_0[(long)(b * NI + i) * DC + c] = acc;
  }
}

// ---------------- Kernel 2: el-el pair features -> h_el_0 ----------------
__global__ __launch_bounds__(128) void k_el_el(
    const float* __restrict__ diff_ee, const float* __restrict__ dist_ee,
    const float* __restrict__ feat_ee,
    const float* __restrict__ sb0, const float* __restrict__ sb1,
    const float* __restrict__ sscale, const float* __restrict__ sw0,
    const float* __restrict__ sw1, const float* __restrict__ swenv,
    const float* __restrict__ Gsame, const float* __restrict__ Gdiff,
    const float* __restrict__ Wsame, const float* __restrict__ bsame,
    const float* __restrict__ Wdiff, const float* __restrict__ bdiff,
    const float* __restrict__ el0_w, const float* __restrict__ el0_b,
    float* __restrict__ h_el_0) {
  __shared__ float Hs[64 * 65];
  __shared__ float Es[64 * 65];
  __shared__ float Ys[64 * 33];
  __shared__ float Ws[64 * 33];
  __shared__ float hfb[2 * 64];
  const int b = blockIdx.x >> 4;
  const int i0 = (blockIdx.x & 15) * 2;
  const int tid = threadIdx.x;
  const int lane = tid & 31, wave = tid >> 5;
  const int half = lane >> 4, idx = lane & 15;

  for (int o = tid; o < 64 * 64; o += 128) {
    int r = o >> 6, c = o & 63;
    int i = i0 + (r >> 5), j = r & 31;
    long p = (long)(b * NE + i) * NE + j;
    const float* d = diff_ee + p * 3;
    float dist = dist_ee[p];
    float h = d[0] * sw0[c] + d[1] * sw0[64 + c] + d[2] * sw0[128 + c] + sb0[c];
    Hs[r * 65 + c] = silu_f(h);
    float sc = softplus_f(sscale[c]);
    float t = dist * sc;
    Es[r * 65 + c] = __expf(-t * t);
  }
  __syncthreads();
  for (int q = 0; q < 4; q++) {
    int job = wave * 4 + q;
    int mt = job >> 2, nt = (job >> 1) & 1, mat = job & 1;
    const float* A = mat ? Es : Hs;
    const float* Bw = mat ? swenv : sw1;
    v8f acc = gemm_ldsA(A, 65, mt * 16, Bw, H1C, nt * 16, H0C, lane);
    float* Dst = mat ? Ws : Ys;
#pragma unroll
    for (int r = 0; r < 8; r++)
      Dst[(mt * 16 + half * 8 + r) * 33 + nt * 16 + idx] = acc[r];
  }
  __syncthreads();
  for (int o = tid; o < 64 * 32; o += 128) {
    int r = o >> 5, c = o & 31;
    Ys[r * 33 + c] = (Ys[r * 33 + c] + sb1[c]) * Ws[r * 33 + c];
  }
  __syncthreads();
  for (int q = 0; q < 2; q++) {
    int job = wave * 2 + q;
    int mt = job >> 1, nt = job & 1;
    int ie = mt >> 1, jt = mt & 1;
    int i = i0 + ie;
    int up_i = (i < NUP) ? 1 : 0;
    int up_j = (jt == 0) ? 1 : 0;
    int same = (up_i == up_j) ? 1 : 0;
    const float* Gam = same ? Gsame : Gdiff;
    const float* W4 = same ? Wsame : Wdiff;
    const float* bia = same ? bsame : bdiff;
    v8f accG = gemm_ldsA(Ys, 33, mt * 16, Gam, DEE, nt * 16, H1C, lane);
    long prow = (long)(b * NE + i) * NE + (jt * 16 + idx);
    float a0 = feat_ee[prow * 4 + 2 * half + 0];
    float a1 = feat_ee[prow * 4 + 2 * half + 1];
    float w0v = W4[(2 * half + 0) * DEE + nt * 16 + idx];
    float w1v = W4[(2 * half + 1) * DEE + nt * 16 + idx];
    v8f accF = {0.f, 0.f, 0.f, 0.f, 0.f, 0.f, 0.f, 0.f};
    accF = wmma4(accF, a0, a1, w0v, w1v);
    float bn = bia[nt * 16 + idx];
    float partial = 0.f;
#pragma unroll
    for (int r = 0; r < 8; r++)
      partial += silu_f(accF[r] + bn) * accG[r];
    partial += __shfl_xor(partial, 16);
    int cat = same ? 0 : 1;
    if (half == 0)
      hfb[ie * 64 + cat * 32 + nt * 16 + idx] = partial;
  }
  __syncthreads();
  for (int o = tid; o < 2 * 64; o += 128) {
    int ie = o >> 6, c = o & 63;
    float acc = el0_b[c];
#pragma unroll 4
    for (int k = 0; k < 64; k++) acc += hfb[ie * 64 + k] * el0_w[k * DC + c];
    h_el_0[(long)(b * NE + i0 + ie) * DC + c] = silu_f(acc);
  }
}

// ---------------- Kernel 3: el-ion messages: h_el_ion_0 + h_ion_up/dn ----------------
__global__ __launch_bounds__(128) void k_el_ion_msg(
    const float* __restrict__ diff_ei, const float* __restrict__ dist_ei,
    const float* __restrict__ feat_ei,
    const float* __restrict__ sb0, const float* __restrict__ sb1,
    const float* __restrict__ sscale, const float* __restrict__ sw0,
    const float* __restrict__ sw1, const float* __restrict__ swenv,
    const float* __restrict__ Gelion, const float* __restrict__ gW,
    const float* __restrict__ h_el_0, const float* __restrict__ h_ion_0,
    float* __restrict__ h_el_ion_0,
    float* __restrict__ h_ion_up, float* __restrict__ h_ion_dn) {
  __shared__ float Hs[16 * 65];
  __shared__ float Es[16 * 65];
  __shared__ float Ys[16 * 33];
  __shared__ float Ws[16 * 33];
  const int b = blockIdx.x >> 4;
  const int e0 = (blockIdx.x & 15) * 2;
  const int tid = threadIdx.x;
  const int lane = tid & 31, wave = tid >> 5;
  const int half = lane >> 4, idx = lane & 15;

  for (int o = tid; o < 16 * 64; o += 128) {
    int r = o >> 6, c = o & 63;
    int e = e0 + (r >> 3), ion = r & 7;
    long p = (long)(b * NE + e) * NI + ion;
    const float* d = diff_ei + p * 3;
    float dist = dist_ei[p];
    float h = d[0] * sw0[c] + d[1] * sw0[64 + c] + d[2] * sw0[128 + c] + sb0[c];
    Hs[r * 65 + c] = silu_f(h);
    float sc = softplus_f(sscale[c]);
    float t = dist * sc;
    Es[r * 65 + c] = __expf(-t * t);
  }
  __syncthreads();
  {
    int mat = wave >> 1, nt = wave & 1;
    const float* A = mat ? Es : Hs;
    const float* Bw = mat ? swenv : sw1;
    v8f acc = gemm_ldsA(A, 65, 0, Bw, H1C, nt * 16, H0C, lane);
    float* Dst = mat ? Ws : Ys;
#pragma unroll
    for (int r = 0; r < 8; r++)
      Dst[(half * 8 + r) * 33 + nt * 16 + idx] = acc[r];
  }
  __syncthreads();
  for (int o = tid; o < 16 * 32; o += 128) {
    int r = o >> 5, c = o & 31;
    Ys[r * 33 + c] = (Ys[r * 33 + c] + sb1[c]) * Ws[r * 33 + c];
  }
  __syncthreads();
  {
    int nt = wave;  // DC = 64 -> 4 N-tiles, one per wave
    v8f accG = gemm_ldsA(Ys, 33, 0, Gelion, DC, nt * 16, H1C, lane);
    long prow = (long)(b * NE + e0 + (idx >> 3)) * NI + (idx & 7);
    float a0 = feat_ei[prow * 4 + 2 * half + 0];
    float a1 = feat_ei[prow * 4 + 2 * half + 1];
    float w0v = gW[(2 * half + 0) * DC + nt * 16 + idx];
    float w1v = gW[(2 * half + 1) * DC + nt * 16 + idx];
    v8f accF = {0.f, 0.f, 0.f, 0.f, 0.f, 0.f, 0.f, 0.f};
    accF = wmma4(accF, a0, a1, w0v, w1v);
    int n = nt * 16 + idx;
    int e = e0 + half;
    float hel = h_el_0[(long)(b * NE + e) * DC + n];
    float* hi_dst = (e0 < NUP) ? h_ion_up : h_ion_dn;
#pragma unroll
    for (int r = 0; r < 8; r++) {
      float h0v = silu_f(accF[r] + hel + h_ion_0[(long)(b * NI + r) * DC + n]);
      h_el_ion_0[((long)(b * NE + e) * NI + r) * DC + n] = h0v;
      float msg = h0v * accG[r];
      msg += __shfl_xor(msg, 16);
      if (half == 0)
        atomicAdd(&hi_dst[(long)(b * NI + r) * DC + n], msg);
    }
  }
}

// ---------------- Kernel 4: ion MLP iterate + h_ion_out ----------------
__global__ __launch_bounds__(128) void k_ion_mlp(
    const float* __restrict__ h_ion_up, const float* __restrict__ h_ion_dn,
    const float* __restrict__ mlp_w, const float* __restrict__ mlp_b,
    const float* __restrict__ lout_ion_w,
    float* __restrict__ h_ion_out) {
  __shared__ float up[8 * 64];
  __shared__ float dn[8 * 64];
  __shared__ float upd[8 * 2 * 64];
  __shared__ float fb[16 * 130];
  const int b = blockIdx.x;
  const int tid = threadIdx.x;
  const int lane = tid & 31, wave = tid >> 5;
  const int half = lane >> 4, idx = lane & 15;

  for (int o = tid; o < 8 * 64; o += 128) {
    up[o] = h_ion_up[(long)b * NI * DC + o];
    dn[o] = h_ion_dn[(long)b * NI * DC + o];
  }
  __syncthreads();
  for (int o = tid; o < 8 * 2 * 64; o += 128) {
    int i = o >> 7, s = (o >> 6) & 1, c = o & 63;
    const float* first = s ? dn : up;
    const float* second = s ? up : dn;
    float acc = mlp_b[c];
#pragma unroll 4
    for (int k = 0; k < 64; k++) acc += first[i * 64 + k] * mlp_w[k * DC + c];
#pragma unroll 4
    for (int k = 0; k < 64; k++) acc += second[i * 64 + k] * mlp_w[(64 + k) * DC + c];
    upd[o] = silu_f(acc);
  }
  __syncthreads();
  for (int o = tid; o < 8 * 64; o += 128) {
    int i = o >> 6, c = o & 63;
    float nu = (upd[(i * 2 + 0) * 64 + c] + up[o]) * 0.7071067811865476f;
    float nd = (upd[(i * 2 + 1) * 64 + c] + dn[o]) * 0.7071067811865476f;
    up[o] = nu;
    dn[o] = nd;
  }
  __syncthreads();
  for (int o = tid; o < 16 * 128; o += 128) {
    int r = o >> 7, k = o & 127;
    int i = r >> 1, s = r & 1;
    const float* first = s ? dn : up;
    const float* second = s ? up : dn;
    fb[r * 130 + k] = (k < 64) ? first[i * 64 + k] : second[i * 64 + k - 64];
  }
  __syncthreads();
  for (int nt = wave; nt < 16; nt += 4) {
    v8f acc = gemm_ldsA(fb, 130, 0, lout_ion_w, DOUT, nt * 16, 128, lane);
#pragma unroll
    for (int r = 0; r < 8; r++) {
      int M = half * 8 + r;
      h_ion_out[(((long)b * NI + (M >> 1)) * 2 + (M & 1)) * DOUT + nt * 16 + idx] = acc[r];
    }
  }
}

// ---------------- Kernel 5: el_out = h_el_0 @ lin_out_el.w ----------------
__global__ __launch_bounds__(128) void k_el_out(
    const float* __restrict__ h_el_0, const float* __restrict__ lout_el_w,
    float* __restrict__ el_out) {
  const int b = blockIdx.x;
  const int tid = threadIdx.x;
  const int lane = tid & 31, wave = tid >> 5;
  const int half = lane >> 4, idx = lane & 15;
  for (int q = 0; q < 8; q++) {
    int job = wave * 8 + q;
    int mt = job >> 4, nt = job & 15;
    const float* A = h_el_0 + (long)(b * NE + mt * 16) * DC;
    v8f acc = gemm_globA(A, DC, lout_el_w, DOUT, nt * 16, DC, lane);
#pragma unroll
    for (int r = 0; r < 8; r++) {
      int M = mt * 16 + half * 8 + r;
      el_out[(long)(b * NE + M) * DOUT + nt * 16 + idx] = acc[r];
    }
  }
}

// ---------------- Kernel 6: final fused output stage ----------------
__global__ __launch_bounds__(128) void k_final(
    const float* __restrict__ diff_ei, const float* __restrict__ dist_ei,
    const float* __restrict__ sb0, const float* __restrict__ sb1,
    const float* __restrict__ sscale, const float* __restrict__ sw0,
    const float* __restrict__ sw1, const float* __restrict__ swenv,
    const float* __restrict__ Gout, const float* __restrict__ Welion,
    const float* __restrict__ belion,
    const float* __restrict__ h_el_ion_0,
    const float* __restrict__ h_ion_out, const float* __restrict__ el_out,
    float* __restrict__ out_h_el, float* __restrict__ out_h_el_ion) {
  __shared__ float Hs[16 * 65];
  __shared__ float Es[16 * 65];
  __shared__ float Ys[16 * 33];
  __shared__ float Ws[16 * 33];
  const int b = blockIdx.x >> 4;
  const int e0 = (blockIdx.x & 15) * 2;
  const int tid = threadIdx.x;
  const int lane = tid & 31, wave = tid >> 5;
  const int half = lane >> 4, idx = lane & 15;

  for (int o = tid; o < 16 * 64; o += 128) {
    int r = o >> 6, c = o & 63;
    int e = e0 + (r >> 3), ion = r & 7;
    long p = (long)(b * NE + e) * NI + ion;
    const float* d = diff_ei + p * 3;
    float dist = dist_ei[p];
    float h = d[0] * sw0[c] + d[1] * sw0[64 + c] + d[2] * sw0[128 + c] + sb0[c];
    Hs[r * 65 + c] = silu_f(h);
    float sc = softplus_f(sscale[c]);
    float t = dist * sc;
    Es[r * 65 + c] = __expf(-t * t);
  }
  __syncthreads();
  {
    int mat = wave >> 1, nt = wave & 1;
    const float* A = mat ? Es : Hs;
    const float* Bw = mat ? swenv : sw1;
    v8f acc = gemm_ldsA(A, 65, 0, Bw, H1C, nt * 16, H0C, lane);
    float* Dst = mat ? Ws : Ys;
#pragma unroll
    for (int r = 0; r < 8; r++)
      Dst[(half * 8 + r) * 33 + nt * 16 + idx] = acc[r];
  }
  __syncthreads();
  for (int o = tid; o < 16 * 32; o += 128) {
    int r = o >> 5, c = o & 31;
    Ys[r * 33 + c] = (Ys[r * 33 + c] + sb1[c]) * Ws[r * 33 + c];
  }
  __syncthreads();
  const float* A0 = h_el_ion_0 + (long)(b * NE + e0) * NI * DC;  // 16 rows, stride DC
  const int e = e0 + half;
  const int s = (e0 < NUP) ? 0 : 1;
  for (int nt = wave; nt < 16; nt += 4) {
    v8f accE = gemm_globA(A0, DC, Welion, DOUT, nt * 16, DC, lane);
    v8f accG = gemm_ldsA(Ys, 33, 0, Gout, DOUT, nt * 16, H1C, lane);
    int n = nt * 16 + idx;
    float bn = belion[n];
    float eo = el_out[(long)(b * NE + e) * DOUT + n];
    float hsum = 0.f;
#pragma unroll
    for (int r = 0; r < 8; r++) {
      float hio = h_ion_out[(((long)b * NI + r) * 2 + s) * DOUT + n];
      float pre = accE[r] + bn + hio + eo;
      float val = silu_f(pre) * accG[r];
      out_h_el_ion[((long)(b * NE + e) * NI + r) * DOUT + n] = val;
      hsum += val;
    }
    out_h_el[(long)(b * NE + e) * DOUT + n] = hsum;
  }
}

extern "C" void kernel_launch(void* const* d_in, const int* in_sizes, int n_in,
                              void* d_out, int out_size, void* d_ws, size_t ws_size,
                              hipStream_t stream) {
  (void)in_sizes; (void)n_in; (void)out_size; (void)ws_size;
  // setup_inputs() insertion order; params pytree-flattened (sorted dict keys)
  const float* diff_ee = (const float*)d_in[0];
  const float* dist_ee = (const float*)d_in[1];
  const float* diff_ei = (const float*)d_in[2];
  const float* dist_ei = (const float*)d_in[3];
  const float* diff_ii = (const float*)d_in[4];
  const float* dist_ii = (const float*)d_in[5];
  const float* feat_ion = (const float*)d_in[6];
  const float* feat_ee = (const float*)d_in[7];
  const float* feat_ei = (const float*)d_in[8];
  const float* Gdiff = (const float*)d_in[9];     // Gamma_diff.w [32,32]
  const float* Gelion = (const float*)d_in[10];   // Gamma_el_ion.w [32,64]
  const float* Gii = (const float*)d_in[11];      // Gamma_ion_ion.w [32,64]
  const float* Gout = (const float*)d_in[12];     // Gamma_out.w [32,256]
  const float* Gsame = (const float*)d_in[13];    // Gamma_same.w [32,32]
  const float* el0_b = (const float*)d_in[14];    // el_0 {b,w}
  const float* el0_w = (const float*)d_in[15];
  const float* eed_b = (const float*)d_in[16];    // el_el_diff {b,w}
  const float* eed_w = (const float*)d_in[17];
  const float* ees_b = (const float*)d_in[18];    // el_el_same {b,w}
  const float* ees_w = (const float*)d_in[19];
  const float* gW = (const float*)d_in[20];       // g_el_ion.w [4,64]
  const float* emb_b = (const float*)d_in[21];    // ion_emb {b,w}
  const float* emb_w = (const float*)d_in[22];
  const float* mlp_b = (const float*)d_in[23];    // ion_mlp_0 {b,w}
  const float* mlp_w = (const float*)d_in[24];
  const float* loutel_w = (const float*)d_in[25];   // lin_out_el.w [64,256]
  const float* loutei_b = (const float*)d_in[26];   // lin_out_elion {b,w}
  const float* loutei_w = (const float*)d_in[27];
  const float* loution_w = (const float*)d_in[28];  // lin_out_ion.w [128,256]
  // sf params: {b0,b1,scale,w0,w1,w_env}
  const float* see_b0 = (const float*)d_in[29];
  const float* see_b1 = (const float*)d_in[30];
  const float* see_sc = (const float*)d_in[31];
  const float* see_w0 = (const float*)d_in[32];
  const float* see_w1 = (const float*)d_in[33];
  const float* see_we = (const float*)d_in[34];
  const float* sei_b0 = (const float*)d_in[35];
  const float* sei_b1 = (const float*)d_in[36];
  const float* sei_sc = (const float*)d_in[37];
  const float* sei_w0 = (const float*)d_in[38];
  const float* sei_w1 = (const float*)d_in[39];
  const float* sei_we = (const float*)d_in[40];
  const float* sii_b0 = (const float*)d_in[41];
  const float* sii_b1 = (const float*)d_in[42];
  const float* sii_sc = (const float*)d_in[43];
  const float* sii_w0 = (const float*)d_in[44];
  const float* sii_w1 = (const float*)d_in[45];
  const float* sii_we = (const float*)d_in[46];
  const float* so_b0 = (const float*)d_in[47];
  const float* so_b1 = (const float*)d_in[48];
  const float* so_sc = (const float*)d_in[49];
  const float* so_w0 = (const float*)d_in[50];
  const float* so_w1 = (const float*)d_in[51];
  const float* so_we = (const float*)d_in[52];
  // d_in[53] = n_up (hardcoded 16)

  float* ws = (float*)d_ws;
  float* h_ion_0 = ws;
  float* h_el_0 = h_ion_0 + (long)BATCH * NI * DC;
  float* h_el_ion_0 = h_el_0 + (long)BATCH * NE * DC;
  float* h_ion_up = h_el_ion_0 + (long)BATCH * NE * NI * DC;
  float* h_ion_dn = h_ion_up + (long)BATCH * NI * DC;
  float* h_ion_out = h_ion_dn + (long)BATCH * NI * DC;
  float* el_out = h_ion_out + (long)BATCH * NI * 2 * DOUT;

  hipMemsetAsync(h_ion_up, 0, (size_t)BATCH * NI * DC * 2 * sizeof(float), stream);

  k_ion_embed<<<BATCH, 128, 0, stream>>>(diff_ii, dist_ii, feat_ion, emb_w, emb_b,
                                         sii_b0, sii_b1, sii_sc, sii_w0, sii_w1, sii_we,
                                         Gii, h_ion_0);
  k_el_el<<<BATCH * 16, 128, 0, stream>>>(diff_ee, dist_ee, feat_ee,
                                          see_b0, see_b1, see_sc, see_w0, see_w1, see_we,
                                          Gsame, Gdiff, ees_w, ees_b, eed_w, eed_b,
                                          el0_w, el0_b, h_el_0);
  k_el_ion_msg<<<BATCH * 16, 128, 0, stream>>>(diff_ei, dist_ei, feat_ei,
                                               sei_b0, sei_b1, sei_sc, sei_w0, sei_w1, sei_we,
                                               Gelion, gW, h_el_0, h_ion_0,
                                               h_el_ion_0, h_ion_up, h_ion_dn);
  k_ion_mlp<<<BATCH, 128, 0, stream>>>(h_ion_up, h_ion_dn, mlp_w, mlp_b, loution_w, h_ion_out);
  k_el_out<<<BATCH, 128, 0, stream>>>(h_el_0, loutel_w, el_out);
  k_final<<<BATCH * 16, 128, 0, stream>>>(diff_ei, dist_ei,
                                          so_b0, so_b1, so_sc, so_w0, so_w1, so_we,
                                          Gout, loutei_w, loutei_b,
                                          h_el_ion_0, h_ion_out, el_out,
                                          (float*)d_out,
                                          (float*)d_out + (long)BATCH * NE * DOUT);
}